// FIRFilter_1924145349140
// MI455X (gfx1250) — compile-verified
//
#include <hip/hip_runtime.h>

typedef float v2f __attribute__((ext_vector_type(2)));
typedef float v8f __attribute__((ext_vector_type(8)));

#define NTAPS   63
#define BATCH   32
#define SAMPLES 2097152
#define TILE    256        // outputs per 16x16 WMMA macro-tile
#define KCH     20         // K-chunks of 4 -> J = 80
#define TPW     32         // tiles per wave
#define WPB     8          // waves per block
#define LDSF    320        // floats of x staged per tile buffer

// ---------------------------------------------------------------------------
// Kernel 1: adaptor  b' = b @ W^T + bias   (32x63, negligible cost)
// ---------------------------------------------------------------------------
__global__ __launch_bounds__(256) void fir_adaptor_kernel(
    const float* __restrict__ b, const float* __restrict__ W,
    const float* __restrict__ bias, float* __restrict__ bp) {
  for (int idx = threadIdx.x; idx < BATCH * NTAPS; idx += 256) {
    const int i = idx / NTAPS, o = idx % NTAPS;
    float acc = bias[o];
    const float* br = b + i * NTAPS;
    const float* wr = W + o * NTAPS;
    #pragma unroll 9
    for (int t = 0; t < NTAPS; ++t) acc += br[t] * wr[t];
    bp[i * 64 + o] = acc;   // stride 64 for alignment
  }
}

// ---------------------------------------------------------------------------
// Async fill: stage 320 floats x[g0 .. g0+319] into LDS (byte base
// ldsByteBase) via 3x global_load_async_to_lds_b128. Per-lane addresses are
// clamped into [0, SAMPLES-4]; edge tiles fix up out-of-band entries later.
// ---------------------------------------------------------------------------
__device__ __forceinline__ void async_fill_tile(const float* __restrict__ xrow,
                                                int g0, int ldsByteBase,
                                                int lane) {
  int e0 = g0 + 4 * lane;        // float4 element index this lane copies
  int e1 = e0 + 128;
  int e2 = e0 + 256;
  e0 = min(max(e0, 0), SAMPLES - 4);
  e1 = min(max(e1, 0), SAMPLES - 4);
  asm volatile("global_load_async_to_lds_b128 %0, %1, off"
               :: "v"(ldsByteBase + 16 * lane), "v"(xrow + e0) : "memory");
  asm volatile("global_load_async_to_lds_b128 %0, %1, off"
               :: "v"(ldsByteBase + 512 + 16 * lane), "v"(xrow + e1) : "memory");
  if (lane < 16) {               // EXEC-masked, still one ASYNCcnt increment
    e2 = min(max(e2, 0), SAMPLES - 4);
    asm volatile("global_load_async_to_lds_b128 %0, %1, off"
                 :: "v"(ldsByteBase + 1024 + 16 * lane), "v"(xrow + e2) : "memory");
  }
}

// ---------------------------------------------------------------------------
// Kernel 2: FIR as Toeplitz GEMM on the f32 WMMA pipe, double-buffered
// async-to-LDS fills, two independent WMMA accumulation chains.
//   Y(16x16) = X(16x80) * H(80x16), Y[m][n] = y[t0 + 16m + n]
//   X[m][j] = x[t0 - 32 + 16m + j]            (staged in LDS)
//   H[j][n] = coef[j - n - 1] in-band else 0  (built once per wave)
// ---------------------------------------------------------------------------
__global__ __launch_bounds__(256) void fir_wmma_kernel(
    const float* __restrict__ x, const float* __restrict__ bp,
    float* __restrict__ y) {
  __shared__ float lds[WPB * 2 * LDSF];
  const int lane   = threadIdx.x & 31;
  const int wslot  = threadIdx.x >> 5;
  const int waveId = blockIdx.x * WPB + wslot;
  const int batch  = waveId >> 8;      // 256 strips per batch row
  const int strip  = waveId & 255;
  const int n      = lane & 15;
  const int hi     = lane >> 4;

  const float* xrow = x + (size_t)batch * SAMPLES;
  float*       yrow = y + (size_t)batch * SAMPLES;
  const float* coef = bp + batch * 64;

  // Banded filter operand H, built once per wave (40 VGPRs).
  // Assumed 32-bit B(4x16) layout: VGPR v -> row K=v (lanes 0-15), K=v+2
  // (lanes 16-31), mirroring the documented A(16x4) layout.
  v2f B[KCH];
  #pragma unroll
  for (int q = 0; q < KCH; ++q) {
    #pragma unroll
    for (int v = 0; v < 2; ++v) {
      const int krow = v + 2 * hi;
      const int cidx = 4 * q + krow - n - 1;
      B[q][v] = (cidx >= 0 && cidx < NTAPS) ? coef[cidx] : 0.0f;
    }
  }

  const int m        = n;        // A row for this lane (ISA A 16x4 layout)
  const int koff     = 2 * hi;   // lanes 0-15 hold K{0,1}; 16-31 hold K{2,3}
  const int ldsBase  = wslot * 2 * LDSF * 4;   // this wave's LDS byte base
  const int t0base   = strip * TPW * TILE;

  // Prologue: fill tile 0 into buffer 0.
  async_fill_tile(xrow, t0base - 32, ldsBase, lane);

  for (int t = 0; t < TPW; ++t) {
    const int t0  = t0base + t * TILE;
    const int g0  = t0 - 32;
    const int buf = t & 1;
    float* cur = lds + wslot * 2 * LDSF + buf * LDSF;

    if (t + 1 < TPW) {
      // Prior ds reads of the other buffer must retire before overwriting it.
      asm volatile("s_wait_dscnt 0x0" ::: "memory");
      async_fill_tile(xrow, g0 + TILE, ldsBase + (1 - buf) * LDSF * 4, lane);
      // ASYNCcnt completes in order; <=3 outstanding => fill(t) is done.
      asm volatile("s_wait_asynccnt 0x3" ::: "memory");
    } else {
      asm volatile("s_wait_asynccnt 0x0" ::: "memory");
    }

    // Boundary tiles: zero LDS entries whose source index is out of range
    // (wave-uniform branch; interior tiles skip entirely).
    if (t0 < 32 || t0 + 288 > SAMPLES) {
      for (int u = lane; u < LDSF; u += 32) {
        const int g = g0 + u;
        if (g < 0 || g >= SAMPLES) cur[u] = 0.0f;
      }
    }

    // Two independent WMMA accumulation chains (even / odd K-chunks).
    v8f c0 = {0.f, 0.f, 0.f, 0.f, 0.f, 0.f, 0.f, 0.f};
    v8f c1 = {0.f, 0.f, 0.f, 0.f, 0.f, 0.f, 0.f, 0.f};
    #pragma unroll
    for (int q = 0; q < KCH; q += 2) {
      const v2f a0 = *(const v2f*)(cur + 16 * m + 4 * q + koff);
      const v2f a1 = *(const v2f*)(cur + 16 * m + 4 * (q + 1) + koff);
      c0 = __builtin_amdgcn_wmma_f32_16x16x4_f32(false, a0, false, B[q],
                                                 (short)0, c0, false, false);
      c1 = __builtin_amdgcn_wmma_f32_16x16x4_f32(false, a1, false, B[q + 1],
                                                 (short)0, c1, false, false);
    }

    // C/D layout: VGPR r -> row M=r (lanes 0-15), M=r+8 (lanes 16-31).
    float* yt = yrow + t0 + 128 * hi + n;
    #pragma unroll
    for (int r = 0; r < 8; ++r) yt[16 * r] = c0[r] + c1[r];
  }
}

// ---------------------------------------------------------------------------
extern "C" void kernel_launch(void* const* d_in, const int* in_sizes, int n_in,
                              void* d_out, int out_size, void* d_ws, size_t ws_size,
                              hipStream_t stream) {
  const float* x    = (const float*)d_in[0];
  const float* b    = (const float*)d_in[1];
  const float* W    = (const float*)d_in[2];
  const float* bias = (const float*)d_in[3];
  float* bp = (float*)d_ws;          // 32*64 floats of adapted coefficients
  float* yo = (float*)d_out;

  fir_adaptor_kernel<<<1, 256, 0, stream>>>(b, W, bias, bp);

  const int totalTiles = BATCH * (SAMPLES / TILE);          // 262144
  const int blocks     = totalTiles / (TPW * WPB);          // 1024
  fir_wmma_kernel<<<blocks, 256, 0, stream>>>(x, bp, yo);
}